// LennardJonesModelWrapper_77833397338753
// MI455X (gfx1250) — compile-verified
//
#include <hip/hip_runtime.h>
#include <math.h>

// ---------------------------------------------------------------------------
// Lennard-Jones neighbor-list kernel for MI455X (gfx1250, wave32).
//
// Bandwidth-bound: ~210 MB of neighbor data @ 23.3 TB/s => ~9us floor.
// Wave-per-atom, lane-per-neighbor. Neighbor rows (256B idx + 768B shifts)
// are double-buffered into wave-private LDS with GLOBAL_LOAD_ASYNC_TO_LDS
// (ASYNCcnt) one atom ahead; global_prefetch_b8 warms L2 two atoms ahead.
// Global energy/virial reduced deterministically: thread-local -> wave shfl
// -> block partial row in d_ws -> single-wave WMMA column-sum finalize.
// ---------------------------------------------------------------------------

#define TPB   256
#define NBLK  2048            // fixed grid => deterministic partial layout
#define WPB   (TPB / 32)      // 8 waves per block (wave32)

typedef __attribute__((ext_vector_type(2))) float v2f;
typedef __attribute__((ext_vector_type(8))) float v8f;

#if defined(__AMDGCN__) && __has_builtin(__builtin_amdgcn_wmma_f32_16x16x4_f32)
#define USE_WMMA_REDUCE 1
#else
#define USE_WMMA_REDUCE 0
#endif

#if defined(__AMDGCN__) && \
    __has_builtin(__builtin_amdgcn_global_load_async_to_lds_b64) && \
    __has_builtin(__builtin_amdgcn_global_load_async_to_lds_b128) && \
    __has_builtin(__builtin_amdgcn_s_wait_asynccnt)
#define USE_ASYNC_LDS 1
// Exact parameter types per the compiler diagnostic:
//   b64 : AS1 'int __vector(2)*' src, AS3 'int __vector(2)*' dst, imm, imm
//   b128: AS1 'int __vector(4)*' src, AS3 'int __vector(4)*' dst, imm, imm
typedef int async_v2i __attribute__((__vector_size__(2 * sizeof(int))));
typedef int async_v4i __attribute__((__vector_size__(4 * sizeof(int))));
typedef __attribute__((address_space(1))) async_v2i* gptr_b64;
typedef __attribute__((address_space(1))) async_v4i* gptr_b128;
typedef __attribute__((address_space(3))) async_v2i* lptr_b64;
typedef __attribute__((address_space(3))) async_v4i* lptr_b128;
#else
#define USE_ASYNC_LDS 0
#endif

struct Cell3 { float m[9]; };

// Per-pair LJ + C2 switch, accumulating force / energy / symmetric virial.
__device__ __forceinline__ void lj_pair_body(
    int j, int sx, int sy, int sz, bool valid,
    const float* __restrict__ pos,
    float xi, float yi, float zi, const Cell3& C,
    float& fx, float& fy, float& fz,
    float& eAcc, float& vxx, float& vyy, float& vzz,
    float& vxy, float& vxz, float& vyz)
{
  const float fsx = (float)sx, fsy = (float)sy, fsz = (float)sz;
  const float shx = fsx * C.m[0] + fsy * C.m[3] + fsz * C.m[6];
  const float shy = fsx * C.m[1] + fsy * C.m[4] + fsz * C.m[7];
  const float shz = fsx * C.m[2] + fsy * C.m[5] + fsz * C.m[8];
  // gather (positions: 2.4 MB, resident in 192 MB L2)
  const float dx = pos[3 * (long)j + 0] + shx - xi;
  const float dy = pos[3 * (long)j + 1] + shy - yi;
  const float dz = pos[3 * (long)j + 2] + shz - zi;
  const float r2 = dx * dx + dy * dy + dz * dz;
  const bool within = valid && (r2 < 36.0f);      // CUTOFF^2
  if (!__any(within)) return;                     // skip wave-dead iterations
  const float rr   = sqrtf(fmaxf(r2, 1e-12f));
  const float rinv = 1.0f / rr;
  const float sr   = 3.4f * rinv;                 // SIGMA/r
  const float sr2  = sr * sr;
  const float sr6  = sr2 * sr2 * sr2;
  const float sr12 = sr6 * sr6;
  const float e    = 4.0f * (sr12 - sr6);                     // EPSILON=1
  const float de   = 4.0f * (6.0f * sr6 - 12.0f * sr12) * rinv;
  float x = rr - 5.0f;                            // (r - r_on)/SW, SW=1
  x = fminf(fmaxf(x, 0.0f), 1.0f);
  const float x2  = x * x;
  const float omx = 1.0f - x;
  const float sw  = 1.0f - x * x2 * (10.0f - 15.0f * x + 6.0f * x2);
  const float dsw = -30.0f * x2 * omx * omx;
  const float esw  = e * sw;
  const float desw = de * sw + e * dsw;
  const float coef = within ? desw * rinv : 0.0f;
  if (within) eAcc += esw;
  const float fpx = coef * dx, fpy = coef * dy, fpz = coef * dz;
  fx += fpx; fy += fpy; fz += fpz;
  vxx += fpx * dx; vyy += fpy * dy; vzz += fpz * dz;
  vxy += fpx * dy; vxz += fpx * dz; vyz += fpy * dz;
}

// Deterministic block partial of the 7 global accumulators -> d_ws row.
__device__ __forceinline__ void block_partial(
    int lane, int wid, float eAcc, float vxx, float vyy, float vzz,
    float vxy, float vxz, float vyz, float* __restrict__ partials)
{
  float vals[7] = { eAcc, vxx, vyy, vzz, vxy, vxz, vyz };
#pragma unroll
  for (int j = 0; j < 7; ++j)
    for (int m = 16; m >= 1; m >>= 1)
      vals[j] += __shfl_xor(vals[j], m, 32);
  __shared__ float red[WPB][8];
  if (lane == 0) {
#pragma unroll
    for (int j = 0; j < 7; ++j) red[wid][j] = vals[j];
  }
  __syncthreads();
  if (threadIdx.x == 0) {
    float* prow = partials + (long)blockIdx.x * 16;
#pragma unroll
    for (int j = 0; j < 7; ++j) {
      float s = 0.f;
      for (int w = 0; w < WPB; ++w) s += red[w][j];
      prow[j] = s;
    }
#pragma unroll
    for (int j = 7; j < 16; ++j) prow[j] = 0.f;   // WMMA sums all 16 cols
  }
}

// ---------------------------------------------------------------------------
// K==64 kernel: async HBM->LDS double-buffered neighbor-row staging.
// Per atom: 256B indices + 768B shifts, staged as b64 + b128 + b64 per lane.
// ---------------------------------------------------------------------------
__global__ __launch_bounds__(TPB) void lj_pair_kernel_k64(
    const float* __restrict__ pos, const float* __restrict__ cell,
    const int* __restrict__ nbr, const int* __restrict__ nnum,
    const int* __restrict__ shifts, int N,
    float* __restrict__ forces, float* __restrict__ partials)
{
  const int lane   = threadIdx.x & 31;
  const int wid    = threadIdx.x >> 5;
  const int gwave  = blockIdx.x * WPB + wid;
  const int nwaves = (int)gridDim.x * WPB;

  Cell3 C;
#pragma unroll
  for (int q = 0; q < 9; ++q) C.m[q] = cell[q];

  float eAcc = 0.f, vxx = 0.f, vyy = 0.f, vzz = 0.f, vxy = 0.f, vxz = 0.f, vyz = 0.f;

#if USE_ASYNC_LDS
  // Wave-private double buffer: [0,256) idx row, [256,1024) shift row.
  __shared__ __align__(16) char stage[WPB][2][1024];
  char* const mystage0 = &stage[wid][0][0];
  char* const mystage1 = &stage[wid][1][0];

  // 3 async transfers cover the 1KB row: b64(256B idx) + b128(512B) + b64(256B).
  auto issue_stage = [&](long a, char* ls) {
    const char* gn = (const char*)(nbr + a * 64L);
    const char* gs = (const char*)(shifts + a * 192L);
    __builtin_amdgcn_global_load_async_to_lds_b64(
        (gptr_b64)(gn + lane * 8), (lptr_b64)(ls + lane * 8), 0, 0);
    __builtin_amdgcn_global_load_async_to_lds_b128(
        (gptr_b128)(gs + lane * 16), (lptr_b128)(ls + 256 + lane * 16), 0, 0);
    __builtin_amdgcn_global_load_async_to_lds_b64(
        (gptr_b64)(gs + 512 + lane * 8), (lptr_b64)(ls + 768 + lane * 8), 0, 0);
  };

  long i = gwave;
  int cur = 0;
  if (i < (long)N) issue_stage(i, mystage0);
  for (; i < (long)N; i += nwaves) {
    // Warm L2 two atoms ahead (global_prefetch_b8): 6x128B shifts + 2x128B idx.
    const long ipref = i + 2L * nwaves;
    if (ipref < (long)N) {
      const char* ps = (const char*)(shifts + ipref * 192L);
      const char* pn = (const char*)(nbr + ipref * 64L);
      if (lane < 6)      __builtin_prefetch(ps + (long)lane * 128, 0, 1);
      else if (lane < 8) __builtin_prefetch(pn + (long)(lane - 6) * 128, 0, 1);
    }
    // Stage next atom, then wait for current buffer (in-order completion:
    // <=3 outstanding means the previous 3 transfers have landed in LDS).
    const long inext = i + nwaves;
    if (inext < (long)N) {
      issue_stage(inext, cur ? mystage0 : mystage1);
      __builtin_amdgcn_s_wait_asynccnt(3);
    } else {
      __builtin_amdgcn_s_wait_asynccnt(0);
    }
    asm volatile("" ::: "memory");   // keep LDS reads below the wait

    const int   nn = nnum[i];
    const float xi = pos[3 * i + 0];
    const float yi = pos[3 * i + 1];
    const float zi = pos[3 * i + 2];
    float fx = 0.f, fy = 0.f, fz = 0.f;

    const char* ls  = cur ? mystage1 : mystage0;
    const int* lidx = (const int*)ls;             // 64 indices
    const int* lsh  = (const int*)(ls + 256);     // 64 x int3 (stride-3: no bank conflicts)
#pragma unroll
    for (int half = 0; half < 2; ++half) {
      const int k = lane + 32 * half;
      lj_pair_body(lidx[k], lsh[3 * k + 0], lsh[3 * k + 1], lsh[3 * k + 2],
                   k < nn, pos, xi, yi, zi, C,
                   fx, fy, fz, eAcc, vxx, vyy, vzz, vxy, vxz, vyz);
    }
    for (int m = 16; m >= 1; m >>= 1) {           // wave32 per-atom reduce
      fx += __shfl_xor(fx, m, 32);
      fy += __shfl_xor(fy, m, 32);
      fz += __shfl_xor(fz, m, 32);
    }
    if (lane == 0) {
      float* f = forces + 3 * i;
      f[0] = fx; f[1] = fy; f[2] = fz;
    }
    cur ^= 1;
  }
#else
  // Fallback (also what the host pass parses): direct coalesced row loads.
  for (long i = gwave; i < (long)N; i += nwaves) {
    const long inext = i + nwaves;
    if (inext < (long)N) {
      const char* pn = (const char*)(nbr + inext * 64L);
      const char* ps = (const char*)(shifts + inext * 192L);
      if (lane < 2)      __builtin_prefetch(pn + (long)lane * 128, 0, 1);
      else if (lane < 8) __builtin_prefetch(ps + (long)(lane - 2) * 128, 0, 1);
    }
    const int   nn = nnum[i];
    const float xi = pos[3 * i + 0];
    const float yi = pos[3 * i + 1];
    const float zi = pos[3 * i + 2];
    float fx = 0.f, fy = 0.f, fz = 0.f;
    const int* nrow = nbr + i * 64L;
    const int* srow = shifts + i * 192L;
#pragma unroll
    for (int half = 0; half < 2; ++half) {
      const int k = lane + 32 * half;
      lj_pair_body(nrow[k], srow[3 * k + 0], srow[3 * k + 1], srow[3 * k + 2],
                   k < nn, pos, xi, yi, zi, C,
                   fx, fy, fz, eAcc, vxx, vyy, vzz, vxy, vxz, vyz);
    }
    for (int m = 16; m >= 1; m >>= 1) {
      fx += __shfl_xor(fx, m, 32);
      fy += __shfl_xor(fy, m, 32);
      fz += __shfl_xor(fz, m, 32);
    }
    if (lane == 0) {
      float* f = forces + 3 * i;
      f[0] = fx; f[1] = fy; f[2] = fz;
    }
  }
#endif

  block_partial(lane, wid, eAcc, vxx, vyy, vzz, vxy, vxz, vyz, partials);
}

// Generic-K kernel (direct loads) — used only if K != 64.
__global__ __launch_bounds__(TPB) void lj_pair_kernel_gen(
    const float* __restrict__ pos, const float* __restrict__ cell,
    const int* __restrict__ nbr, const int* __restrict__ nnum,
    const int* __restrict__ shifts, int N, int K,
    float* __restrict__ forces, float* __restrict__ partials)
{
  const int lane   = threadIdx.x & 31;
  const int wid    = threadIdx.x >> 5;
  const int gwave  = blockIdx.x * WPB + wid;
  const int nwaves = (int)gridDim.x * WPB;

  Cell3 C;
#pragma unroll
  for (int q = 0; q < 9; ++q) C.m[q] = cell[q];

  float eAcc = 0.f, vxx = 0.f, vyy = 0.f, vzz = 0.f, vxy = 0.f, vxz = 0.f, vyz = 0.f;

  for (long i = gwave; i < (long)N; i += nwaves) {
    const int   nn = nnum[i];
    const float xi = pos[3 * i + 0];
    const float yi = pos[3 * i + 1];
    const float zi = pos[3 * i + 2];
    float fx = 0.f, fy = 0.f, fz = 0.f;
    const int* nrow = nbr + i * (long)K;
    const int* srow = shifts + i * (long)K * 3;
    for (int k = lane; k < K; k += 32) {
      lj_pair_body(nrow[k], srow[3 * k + 0], srow[3 * k + 1], srow[3 * k + 2],
                   k < nn, pos, xi, yi, zi, C,
                   fx, fy, fz, eAcc, vxx, vyy, vzz, vxy, vxz, vyz);
    }
    for (int m = 16; m >= 1; m >>= 1) {
      fx += __shfl_xor(fx, m, 32);
      fy += __shfl_xor(fy, m, 32);
      fz += __shfl_xor(fz, m, 32);
    }
    if (lane == 0) {
      float* f = forces + 3 * i;
      f[0] = fx; f[1] = fy; f[2] = fz;
    }
  }

  block_partial(lane, wid, eAcc, vxx, vyy, vzz, vxy, vxz, vyz, partials);
}

// ---------------------------------------------------------------------------
// Finalize: column-sum the [NBLK,16] partials with V_WMMA_F32_16X16X4_F32.
// Partials ride in A (documented 16x4 f32 layout); B = all-ones so the
// result is invariant to any K-slot permutation. Lane 0 reads the 7 sums
// directly from the documented C/D layout (VGPR r <-> M=r).
// ---------------------------------------------------------------------------
__global__ __launch_bounds__(32) void lj_finalize_kernel(
    const float* __restrict__ partials,
    const float* __restrict__ cell,
    float* __restrict__ out, int nrows, int out_size)
{
  const int lane = threadIdx.x;
  float r0, r1, r2s, r3, r4, r5, r6;

#if USE_WMMA_REDUCE
  v8f acc = {0.f, 0.f, 0.f, 0.f, 0.f, 0.f, 0.f, 0.f};
  v2f ones; ones.x = 1.0f; ones.y = 1.0f;
  const int col   = lane & 15;               // M = output slot
  const int rbase = (lane < 16) ? 0 : 2;     // VGPR0 holds K=0 (lo) / K=2 (hi)
  for (int b = 0; b < nrows; b += 4) {       // 4 partial rows per WMMA
    v2f a;
    a.x = partials[(long)(b + rbase) * 16 + col];
    a.y = partials[(long)(b + rbase + 1) * 16 + col];
    acc = __builtin_amdgcn_wmma_f32_16x16x4_f32(
        false, a, false, ones, (short)0, acc, false, false);
  }
  r0 = acc[0]; r1 = acc[1]; r2s = acc[2]; r3 = acc[3];
  r4 = acc[4]; r5 = acc[5]; r6 = acc[6];
#else
  const int col = lane & 15;
  float colsum = 0.f;
  for (int b = 0; b < nrows; ++b) colsum += partials[(long)b * 16 + col];
  r0  = __shfl(colsum, 0, 32); r1 = __shfl(colsum, 1, 32);
  r2s = __shfl(colsum, 2, 32); r3 = __shfl(colsum, 3, 32);
  r4  = __shfl(colsum, 4, 32); r5 = __shfl(colsum, 5, 32);
  r6  = __shfl(colsum, 6, 32);
#endif

  const float c00 = cell[0], c01 = cell[1], c02 = cell[2];
  const float c10 = cell[3], c11 = cell[4], c12 = cell[5];
  const float c20 = cell[6], c21 = cell[7], c22 = cell[8];
  const float det = c00 * (c11 * c22 - c12 * c21)
                  - c01 * (c10 * c22 - c12 * c20)
                  + c02 * (c10 * c21 - c11 * c20);
  const float vol = fabsf(det);

  if (lane == 0) {
    out[0] = 0.5f * r0;                       // energy (half-counted pairs)
    const float sfac = -0.5f / vol;           // stress = -0.5*W_sum/volume
    float* st = out + (out_size - 9);         // row-major 3x3, symmetric
    st[0] = r1 * sfac; st[1] = r4 * sfac; st[2] = r5 * sfac;
    st[3] = r4 * sfac; st[4] = r2s * sfac; st[5] = r6 * sfac;
    st[6] = r5 * sfac; st[7] = r6 * sfac; st[8] = r3 * sfac;
  }
}

extern "C" void kernel_launch(void* const* d_in, const int* in_sizes, int n_in,
                              void* d_out, int out_size, void* d_ws, size_t ws_size,
                              hipStream_t stream) {
  const float* pos    = (const float*)d_in[0];
  const float* cell   = (const float*)d_in[1];
  const int*   nbr    = (const int*)d_in[2];
  const int*   nnum   = (const int*)d_in[3];
  const int*   shifts = (const int*)d_in[4];
  // d_in[5] (batch) unused: single graph, batch == 0.

  const int N = in_sizes[3];            // num_neighbors has N elements
  const int K = in_sizes[2] / N;        // neighbor_matrix is [N,K]

  float* out      = (float*)d_out;      // [energy(1) | forces(3N) | stress(9)]
  float* partials = (float*)d_ws;       // NBLK x 16 floats = 128 KB

  if (K == 64) {
    lj_pair_kernel_k64<<<NBLK, TPB, 0, stream>>>(pos, cell, nbr, nnum, shifts,
                                                 N, out + 1, partials);
  } else {
    lj_pair_kernel_gen<<<NBLK, TPB, 0, stream>>>(pos, cell, nbr, nnum, shifts,
                                                 N, K, out + 1, partials);
  }
  lj_finalize_kernel<<<1, 32, 0, stream>>>(partials, cell, out, NBLK, out_size);
}